// MLayer_70583492542946
// MI455X (gfx1250) — compile-verified
//
#include <hip/hip_runtime.h>

// ---------------------------------------------------------------------------
// Matrix-exponential layer for MI455X (gfx1250), all-f32 via V_WMMA_F32_16X16X4_F32
//   kernel 1: mat0[b] = (sum_a x[b,a] W[a] + bias) / 256 + I      (GEMM, WMMA)
//   kernel 2: mat = mat^2, 8 times (32x32 per batch, WMMA, LDS-staged)
// ---------------------------------------------------------------------------

typedef __attribute__((ext_vector_type(2))) float v2f;
typedef __attribute__((ext_vector_type(8))) float v8f;

#define B_TOTAL 131072
#define SCALE   (1.0f / 256.0f)

static __device__ __forceinline__ v8f wmma4(v2f a, v2f b, v8f c) {
  // D = A(16x4,f32) * B(4x16,f32) + C(16x16,f32)
  return __builtin_amdgcn_wmma_f32_16x16x4_f32(
      /*neg_a=*/false, a, /*neg_b=*/false, b,
      /*c_mod=*/(short)0, c, /*reuse_a=*/false, /*reuse_b=*/false);
}

// ============================================================================
// Kernel 1: generator GEMM.  C[b, mn] = sum_a X[b,a] * W2[a, mn]
//   block = 256 threads (8 waves), 128 batches per block.
//   X staged in LDS with row stride 132 floats (bank-conflict-free A gathers).
//   Each wave handles mn-tiles {wv, wv+8, ...}; 8 batch sub-tiles share each
//   B chunk (each W2 load feeds 8 WMMAs; W2 L2 traffic = 512 MB total).
//   Scale + bias + identity fused into the store.
// ============================================================================
__global__ __launch_bounds__(256) void gen_kernel(const float* __restrict__ x,
                                                  const float* __restrict__ w2,
                                                  const float* __restrict__ bias,
                                                  float* __restrict__ out) {
  __shared__ float xs[128 * 132];  // 67,584 B (gfx1250: 320 KB/WGP)
  const int t  = threadIdx.x;
  const int b0 = blockIdx.x * 128;

  // stage X[128][128] -> xs (padded stride 132)
#pragma unroll
  for (int i = 0; i < 16; ++i) {
    const int f = i * 1024 + t * 4;           // 0..16383
    const float4 v = *(const float4*)(x + b0 * 128 + f);
    const int bl = f >> 7, k = f & 127;
    *(float4*)&xs[bl * 132 + k] = v;
  }
  __syncthreads();

  const int wv = t >> 5, lane = t & 31, lo = lane & 15, hi = lane >> 4;

  for (int tile = wv; tile < 64; tile += 8) {
    const int mn = tile * 16 + lo;            // output column (0..1023)
    const int m = mn >> 5, n = mn & 31;
    const float badd = bias[mn] * SCALE + ((m == n) ? 1.0f : 0.0f);

    v8f C[8];
#pragma unroll
    for (int bb = 0; bb < 8; ++bb) C[bb] = {};

    const float* wp = w2 + hi * 2048 + mn;    // B base: row (2*hi), col mn
    const float* xp = &xs[lo * 132 + 2 * hi]; // A base: row lo, k = 2*hi

#pragma unroll 4
    for (int c = 0; c < 32; ++c) {            // K chunks of 4 (K = 128)
      v2f bv;
      bv.x = wp[c * 4096];                    // W2[4c+2hi      ][mn]
      bv.y = wp[c * 4096 + 1024];             // W2[4c+2hi + 1  ][mn]
#pragma unroll
      for (int bb = 0; bb < 8; ++bb) {
        const v2f av = *(const v2f*)(xp + bb * 2112 + c * 4);  // bb*16*132
        C[bb] = wmma4(av, bv, C[bb]);
      }
    }

    // C layout: VGPR r = batch row (r + 8*hi) of sub-tile bb, column mn
#pragma unroll
    for (int bb = 0; bb < 8; ++bb) {
      float* op = out + (b0 + bb * 16 + 8 * hi) * 1024 + mn;
#pragma unroll
      for (int r = 0; r < 8; ++r)
        op[r * 1024] = C[bb][r] * SCALE + badd;
    }
  }
}

// ============================================================================
// Kernel 2: 8 repeated squarings of each 32x32 matrix (in-place on `out`).
//   block = 128 threads (4 waves); each wave owns 2 matrices (ILP).
//   LDS image padded: row stride 42 floats, matrix stride 1346 floats
//   (bank-conflict-free for A float2 loads, B dword loads, C stores).
//   Squaring 1: global -> regs; squarings 2..7: LDS; squaring 8 -> global.
// ============================================================================
__global__ __launch_bounds__(128) void sq_kernel(float* __restrict__ out) {
  __shared__ float ms[4 * 2 * 1346];          // 43,072 B
  const int t = threadIdx.x;
  const int wv = t >> 5, lane = t & 31, lo = lane & 15, hi = lane >> 4;
  const int b0 = (blockIdx.x * 4 + wv) * 2;   // first of this wave's 2 batches
  const int mbase = wv * (2 * 1346);

  v8f C[2][4];                                // [pair][tile i*2+j]

  // ---- squaring #1: operands gathered from global ----
#pragma unroll
  for (int p = 0; p < 2; ++p) {
    const float* g  = out + (b0 + p) * 1024;
    const float* ga = g + lo * 32 + 2 * hi;   // A: row lo, k base 2*hi
    const float* gb = g + hi * 64 + lo;       // B: row 2*hi, col lo
    v2f Aa[2][8], Bb[2][8];
#pragma unroll
    for (int i = 0; i < 2; ++i)
#pragma unroll
      for (int c = 0; c < 8; ++c)
        Aa[i][c] = *(const v2f*)(ga + i * 512 + c * 4);
#pragma unroll
    for (int j = 0; j < 2; ++j)
#pragma unroll
      for (int c = 0; c < 8; ++c) {
        v2f bv;
        bv.x = gb[c * 128 + j * 16];
        bv.y = gb[c * 128 + j * 16 + 32];
        Bb[j][c] = bv;
      }
#pragma unroll
    for (int i = 0; i < 2; ++i)
#pragma unroll
      for (int j = 0; j < 2; ++j) {
        v8f acc = {};
#pragma unroll
        for (int c = 0; c < 8; ++c) acc = wmma4(Aa[i][c], Bb[j][c], acc);
        C[p][i * 2 + j] = acc;
      }
  }

  // store C regs -> padded LDS image of matrix p
  auto storeLDS = [&](int p) {
    float* s = &ms[mbase + p * 1346 + hi * 336 + lo];  // 8*hi rows = 336
#pragma unroll
    for (int i = 0; i < 2; ++i)
#pragma unroll
      for (int j = 0; j < 2; ++j)
#pragma unroll
        for (int r = 0; r < 8; ++r)
          s[i * 672 + r * 42 + j * 16] = C[p][i * 2 + j][r];
  };

  // one squaring with operands from LDS image of matrix p -> C regs
  auto squareLDS = [&](int p) {
    const float* sa = &ms[mbase + p * 1346 + lo * 42 + 2 * hi];
    const float* sb = &ms[mbase + p * 1346 + hi * 84 + lo];
    v2f Aa[2][8], Bb[2][8];
#pragma unroll
    for (int i = 0; i < 2; ++i)
#pragma unroll
      for (int c = 0; c < 8; ++c)
        Aa[i][c] = *(const v2f*)(sa + i * 672 + c * 4);
#pragma unroll
    for (int j = 0; j < 2; ++j)
#pragma unroll
      for (int c = 0; c < 8; ++c) {
        v2f bv;
        bv.x = sb[c * 168 + j * 16];
        bv.y = sb[c * 168 + j * 16 + 42];
        Bb[j][c] = bv;
      }
#pragma unroll
    for (int i = 0; i < 2; ++i)
#pragma unroll
      for (int j = 0; j < 2; ++j) {
        v8f acc = {};
#pragma unroll
        for (int c = 0; c < 8; ++c) acc = wmma4(Aa[i][c], Bb[j][c], acc);
        C[p][i * 2 + j] = acc;
      }
  };

  storeLDS(0); storeLDS(1);

#pragma unroll 1
  for (int it = 0; it < 6; ++it) {            // squarings #2..#7
    squareLDS(0); squareLDS(1);
    storeLDS(0);  storeLDS(1);
  }

  // ---- squaring #8: operands from LDS, result to global ----
  squareLDS(0); squareLDS(1);
#pragma unroll
  for (int p = 0; p < 2; ++p) {
    float* g = out + (b0 + p) * 1024 + hi * 256 + lo;  // 8*hi rows = 256
#pragma unroll
    for (int i = 0; i < 2; ++i)
#pragma unroll
      for (int j = 0; j < 2; ++j)
#pragma unroll
        for (int r = 0; r < 8; ++r)
          g[i * 512 + r * 32 + j * 16] = C[p][i * 2 + j][r];
  }
}

// ============================================================================
extern "C" void kernel_launch(void* const* d_in, const int* in_sizes, int n_in,
                              void* d_out, int out_size, void* d_ws, size_t ws_size,
                              hipStream_t stream) {
  (void)in_sizes; (void)n_in; (void)out_size; (void)d_ws; (void)ws_size;
  const float* x    = (const float*)d_in[0];   // [131072, 128]
  const float* w2   = (const float*)d_in[1];   // [128, 1024]
  const float* bias = (const float*)d_in[2];   // [1024]
  float* out = (float*)d_out;                  // [131072, 1024]

  gen_kernel<<<B_TOTAL / 128, 256, 0, stream>>>(x, w2, bias, out);  // 1024 blocks
  sq_kernel<<<B_TOTAL / 16, 128, 0, stream>>>(out);                 // 8192 blocks
}